// diff_pred_79336635892364
// MI455X (gfx1250) — compile-verified
//
#include <hip/hip_runtime.h>
#include <math.h>

#define FRAME  160
#define NTAPS  16
#define TLEN   2400
#define NFRAMES 15              // 2400 / 160
#define HALO   16               // need 15 past samples; pad to 16
#define STAGE  (FRAME + HALO)   // 176 staged samples per wave
#define WPB    8                // waves per block (256 threads)

typedef __attribute__((ext_vector_type(2))) float v2f;
typedef __attribute__((ext_vector_type(8))) float v8f;

// u2l: sign(u-128) * (32768/255) * (exp((|u-128|/128)*ln256) - 1)
//    = sign * (32768/255) * (exp2(|u-128|/16) - 1)
__device__ __forceinline__ float u2l(float u) {
    u -= 128.0f;
    float a = fabsf(u);
    float m = (32768.0f / 255.0f) * (__builtin_exp2f(a * (1.0f / 16.0f)) - 1.0f);
    return copysignf(m, u);      // m==0 when u==0, so sign(0)=0 case is exact
}

// l2u: clamp(128 + sign(x) * 128*log1p(|x|*255/32768)/ln256, 0, 255)
//    = clamp(128 + sign(x) * 16*log2(1 + |x|*255/32768), 0, 255)
__device__ __forceinline__ float l2u(float x) {
    float a = fabsf(x);
    float u = 16.0f * __builtin_log2f(fmaf(255.0f / 32768.0f, a, 1.0f));
    u = copysignf(u, x);         // u==0 when x==0 -> result 128, matches sign(0)=0
    return fminf(fmaxf(128.0f + u, 0.0f), 255.0f);
}

__global__ __launch_bounds__(256) void lpc_wmma_kernel(
    const float* __restrict__ sig,   // (2048, 2400)
    const float* __restrict__ lpc,   // (2048, 15, 16)
    float* __restrict__ out,         // (2048, 2400)
    int njobs)                       // 2048*15
{
    __shared__ float xbuf[WPB * STAGE];

    const int lane = threadIdx.x & 31;
    const int wid  = threadIdx.x >> 5;
    const int job  = blockIdx.x * WPB + wid;
    if (job >= njobs) return;        // wave-uniform; terminated waves satisfy barrier

    const int batch = job / NFRAMES;
    const int frame = job - batch * NFRAMES;
    const int tf0   = frame * FRAME;

    float* xw = &xbuf[wid * STAGE];

    // Stage mu-law-decoded signal for this frame plus 16-sample halo:
    // xw[i] = u2l(sig[batch, tf0 - 16 + i]), zeros before t=0.
    for (int i = lane; i < STAGE; i += 32) {
        int t = tf0 - HALO + i;
        float v = 0.0f;
        if (t >= 0) v = u2l(sig[batch * TLEN + t]);
        xw[i] = v;
    }
    __syncthreads();

    const int h = lane >> 4;     // lane half
    const int m = lane & 15;     // M row within tile

    // B chunks: B[k][n] = -lpc[batch,frame,k], replicated across N.
    // 32-bit 4x16 layout: VGPR v, lane-half h -> k = 4*kk + v + 2*h.
    const float* c = &lpc[(batch * NFRAMES + frame) * NTAPS];
    v2f bv[4];
#pragma unroll
    for (int kk = 0; kk < 4; ++kk) {
        int k0 = 4 * kk + 2 * h;
        bv[kk].x = -c[k0];
        bv[kk].y = -c[k0 + 1];
    }

    // 10 tiles of 16 outputs; A[m][k] = xt[t0 + m - k] from LDS.
    for (int ti = 0; ti < FRAME / 16; ++ti) {
        const int base = 16 * ti + HALO + m;   // xw index of xt[t0 + m]
        v8f acc = {};
#pragma unroll
        for (int kk = 0; kk < 4; ++kk) {
            int k0 = 4 * kk + 2 * h;
            v2f a;
            a.x = xw[base - k0];
            a.y = xw[base - (k0 + 1)];
            // D = A x B + C, f32 16x16x4, accumulate K=16 over 4 chunks
            acc = __builtin_amdgcn_wmma_f32_16x16x4_f32(
                false, a, false, bv[kk], (short)0, acc, false, false);
        }
        // D columns are identical (B broadcast over N). VGPR r holds
        // M=r for lanes 0-15 and M=r+8 for lanes 16-31: 16 lanes with
        // (lane&15) < 8 each own one distinct output row.
        if (m < 8) {
            int r = lane & 7;
            float val = acc[0];
#pragma unroll
            for (int q = 1; q < 8; ++q) val = (r == q) ? acc[q] : val;
            int mt = r + 8 * h;
            out[batch * TLEN + tf0 + 16 * ti + mt] = l2u(val);
        }
    }
}

extern "C" void kernel_launch(void* const* d_in, const int* in_sizes, int n_in,
                              void* d_out, int out_size, void* d_ws, size_t ws_size,
                              hipStream_t stream) {
    const float* sig = (const float*)d_in[0];   // (B, 2400, 1) f32
    const float* lpc = (const float*)d_in[1];   // (B, 15, 16)  f32
    float* out = (float*)d_out;                 // (B, 2400, 1) f32

    const int nbatch = in_sizes[0] / TLEN;      // 2048
    const int njobs  = nbatch * NFRAMES;        // 30720
    const int nblocks = (njobs + WPB - 1) / WPB; // 3840

    lpc_wmma_kernel<<<nblocks, 256, 0, stream>>>(sig, lpc, out, njobs);
}